// Embedding_29351806501632
// MI455X (gfx1250) — compile-verified
//
#include <hip/hip_runtime.h>

// ---------------------------------------------------------------------------
// Embedding gather: out[t, :] = W.T[x[t]] + b
//   W: [E=256, V=100000] f32 row-major  -> naive lookup is a 4B-granular
//   column gather (stride 400KB). Roofline: ~162MB compulsory traffic ~ 7us
//   at 23.3 TB/s; the direct gather is request-rate bound (16.7M 4B reqs).
// Strategy:
//   (1) TDM-transpose W into WT[V,256] in d_ws (bias folded), with a
//       double-buffered tensor_load_to_lds pipeline (s_wait_tensorcnt 1).
//   (2) Per-token contiguous 1KB row copies, b128 coalesced, non-temporal
//       stores so the 64MB output stream does not evict WT from L2 (192MB).
// WMMA is intentionally unused: the one-hot GEMM formulation is 3.36 PFLOP
// of wasted work for a zero-FLOP gather.
// ---------------------------------------------------------------------------

typedef __attribute__((ext_vector_type(4))) unsigned int u32x4;
typedef __attribute__((ext_vector_type(8))) int          s32x8;
typedef __attribute__((ext_vector_type(4))) int          s32x4;
typedef __attribute__((ext_vector_type(4))) float        f32x4;

#define TDM_OK (__has_builtin(__builtin_amdgcn_tensor_load_to_lds) && \
                __has_builtin(__builtin_amdgcn_s_wait_tensorcnt))

#define XP_NITER 5   // vocab tiles per block (3125 tiles = 625 blocks * 5)

// ---------------------------------------------------------------------------
// Kernel 1: transpose + bias fold.
// Tile = 32 (vocab, contiguous in W) x 128 (embed); grid.y = 2 covers E=256.
// TDM pad: +1 dword every 32 dwords -> LDS pitch 33 (odd => conflict-free
// transposed reads across 64 banks). Two LDS buffers, software-pipelined DMA.
// ---------------------------------------------------------------------------
__global__ __launch_bounds__(256) void transpose_bias_kernel(
    const float* __restrict__ W, const float* __restrict__ bias,
    float* __restrict__ WT, int V, int ntiles)
{
    constexpr int TV    = 32;        // tile extent along vocab (X / contiguous)
    constexpr int TE    = 128;       // tile extent along embed (Y / strided)
    constexpr int E     = 256;       // full embed width
    constexpr int PITCH = TV + 1;    // dwords per LDS row (via TDM pad)

    __shared__ float tile[2][TE * PITCH];   // 2 * 16896 B = 33792 B

    const int tile0 = blockIdx.x * XP_NITER;          // first vocab tile
    const int e0    = blockIdx.y * TE;                // embed half
    const int tmax  = (ntiles - tile0 < XP_NITER) ? (ntiles - tile0) : XP_NITER;
    if (tmax <= 0) return;

    // Per-thread output mapping: c = embed column within half, two r-groups.
    const int c    = threadIdx.x & (TE - 1);          // 0..127
    const int rgrp = threadIdx.x >> 7;                // 0..1
    const float bv = bias[e0 + c];

#if TDM_OK
    // ---- TDM issue helper (wave 0 only; EXEC is ignored by TDM) ----------
    auto issue = [&](int it, int buf) {
        const int v0 = (tile0 + it) * TV;
        const unsigned long long gaddr =
            (unsigned long long)(const void*)(W + (size_t)e0 * V + (size_t)v0);
        const unsigned lds_off =
            (unsigned)(unsigned long long)(const void*)&tile[buf][0];

        u32x4 g0 = {                       // count=1 | lds | global | type=2
            1u, lds_off, (unsigned)gaddr,
            (unsigned)(gaddr >> 32) | 0x80000000u
        };
        const unsigned td0 = (unsigned)V;  // tensor_dim0 (vocab)
        const unsigned td1 = (unsigned)E;  // tensor_dim1 (embed, full)
        s32x8 g1 = {
            (int)((2u << 16) | (1u << 20) | (4u << 22)),   // 4B | pad 1dw/32dw
            (int)((td0 & 0xFFFFu) << 16),                  // dim0[15:0]
            (int)((td0 >> 16) | ((td1 & 0xFFFFu) << 16)),  // dim0 hi | dim1 lo
            (int)((td1 >> 16) | ((unsigned)TV << 16)),     // dim1 hi | tile_dim0
            (int)TE,                                       // tile_dim1 | tile_dim2=0
            (int)td0,                                      // dim0_stride = V
            0, 0
        };
        s32x4 gz = {0, 0, 0, 0};                           // 2D: groups 2/3 unused
#if __clang_major__ >= 23
        s32x8 g4 = {0, 0, 0, 0, 0, 0, 0, 0};
        __builtin_amdgcn_tensor_load_to_lds(g0, g1, gz, gz, g4, 0);
#else
        __builtin_amdgcn_tensor_load_to_lds(g0, g1, gz, gz, 0);
#endif
    };

    if (threadIdx.x < 32) issue(0, 0);

    for (int i = 0; i < tmax; ++i) {
        const int buf = i & 1;
        if (threadIdx.x < 32) {
            if (i + 1 < tmax) {
                issue(i + 1, buf ^ 1);                    // prefetch next tile
                __builtin_amdgcn_s_wait_tensorcnt(1);     // in-order => tile i done
            } else {
                __builtin_amdgcn_s_wait_tensorcnt(0);
            }
        }
        __syncthreads();                                  // tile i visible to all

        const int v0 = (tile0 + i) * TV;
        float* __restrict__ dst = WT + (size_t)v0 * E + e0 + c;
        #pragma unroll
        for (int j = 0; j < TV / 2; ++j) {
            const int r = rgrp + 2 * j;
            dst[(size_t)r * E] = tile[buf][c * PITCH + r] + bv;
        }
        __syncthreads();       // buf may be overwritten by DMA issued next iter
    }
#else
    // ---- Fallback: cooperative coalesced tile load (no TDM builtin) -------
    const int vv  = threadIdx.x & 31;   // vocab lane (128B coalesced rows)
    const int ee0 = threadIdx.x >> 5;   // 0..7
    for (int i = 0; i < tmax; ++i) {
        const int buf = i & 1;
        const int v0  = (tile0 + i) * TV;
        #pragma unroll
        for (int j = 0; j < TE / 8; ++j) {
            const int ee = ee0 + 8 * j;
            tile[buf][ee * PITCH + vv] =
                W[(size_t)(e0 + ee) * V + (size_t)(v0 + vv)];
        }
        __syncthreads();
        float* __restrict__ dst = WT + (size_t)v0 * E + e0 + c;
        #pragma unroll
        for (int j = 0; j < TV / 2; ++j) {
            const int r = rgrp + 2 * j;
            dst[(size_t)r * E] = tile[buf][c * PITCH + r] + bv;
        }
        __syncthreads();
    }
#endif
}

// ---------------------------------------------------------------------------
// Kernel 2: token row gather from WT (bias already folded).
// One wave handles 4 tokens; per token 2 x b128 coalesced loads and 2 x b128
// NON-TEMPORAL stores (keep the 64MB output stream from evicting WT in L2),
// plus global_prefetch of the next token's 1KB row.
// ---------------------------------------------------------------------------
__global__ __launch_bounds__(256) void gather_rows_kernel(
    const int* __restrict__ idx, const float* __restrict__ WT,
    float* __restrict__ out, int n_tokens)
{
    const int lane = threadIdx.x & 31;
    const int wave = (blockIdx.x * (blockDim.x >> 5)) + (threadIdx.x >> 5);
    int token = wave * 4;

    #pragma unroll
    for (int t = 0; t < 4; ++t, ++token) {
        if (token >= n_tokens) return;
        const int row = idx[token];
        const f32x4* __restrict__ src = (const f32x4*)(WT  + (size_t)row   * 256);
        f32x4*       __restrict__ dst = (f32x4*)      (out + (size_t)token * 256);

        if (t < 3 && token + 1 < n_tokens && lane < 8) {
            const char* nsrc = (const char*)(WT + (size_t)idx[token + 1] * 256);
            __builtin_prefetch(nsrc + lane * 128, 0, 0);   // global_prefetch_b8
        }

        const f32x4 a = src[lane];
        const f32x4 c = src[lane + 32];
        __builtin_nontemporal_store(a, dst + lane);
        __builtin_nontemporal_store(c, dst + lane + 32);
    }
}

// ---------------------------------------------------------------------------
// Fallback: direct strided column gather (only if d_ws too small / odd shape).
// ---------------------------------------------------------------------------
__global__ __launch_bounds__(256) void gather_direct_kernel(
    const int* __restrict__ idx, const float* __restrict__ W,
    const float* __restrict__ bias, float* __restrict__ out,
    int n_tokens, int V, int E)
{
    const int t = blockIdx.x;
    if (t >= n_tokens) return;
    const int row = idx[t];
    for (int e = threadIdx.x; e < E; e += blockDim.x) {
        out[(size_t)t * E + e] = W[(size_t)e * V + row] + bias[e];
    }
}

// ---------------------------------------------------------------------------
extern "C" void kernel_launch(void* const* d_in, const int* in_sizes, int n_in,
                              void* d_out, int out_size, void* d_ws, size_t ws_size,
                              hipStream_t stream)
{
    const int*   x = (const int*)  d_in[0];   // [B*S] token ids
    const float* W = (const float*)d_in[1];   // [E, V] f32
    const float* b = (const float*)d_in[2];   // [E]    f32
    float* out = (float*)d_out;               // [B*S, E] f32

    const int n_tokens = in_sizes[0];
    const int E        = in_sizes[2];
    const int V        = (E > 0) ? in_sizes[1] / E : 0;
    const size_t wt_bytes = (size_t)V * (size_t)E * sizeof(float);

    if (E == 256 && V > 0 && (V % 32) == 0 && ws_size >= wt_bytes) {
        float* WT = (float*)d_ws;                       // [V, 256], bias folded
        const int ntiles = V / 32;                      // 3125
        const int xblks  = (ntiles + XP_NITER - 1) / XP_NITER;  // 625
        transpose_bias_kernel<<<dim3(xblks, 2), 256, 0, stream>>>(W, b, WT,
                                                                  V, ntiles);

        const int tokens_per_block = 8 /*waves*/ * 4 /*tokens/wave*/;
        const int blocks = (n_tokens + tokens_per_block - 1) / tokens_per_block;
        gather_rows_kernel<<<blocks, 256, 0, stream>>>(x, WT, out, n_tokens);
    } else {
        gather_direct_kernel<<<n_tokens, 256, 0, stream>>>(x, W, b, out,
                                                           n_tokens, V, E);
    }
}